// RotatedIoU3DLoss_24309514895594
// MI455X (gfx1250) — compile-verified
//
#include <hip/hip_runtime.h>
#include <stdint.h>

#define TPB 256

typedef float v2f __attribute__((ext_vector_type(2)));
typedef float v8f __attribute__((ext_vector_type(8)));

__device__ __forceinline__ void bev_corners(float x, float y, float w, float l, float a,
                                            float* cx, float* cy) {
  float si, co;
  sincosf(a, &si, &co);
  const float dx[4] = {0.5f, -0.5f, -0.5f, 0.5f};
  const float dy[4] = {0.5f, 0.5f, -0.5f, -0.5f};
#pragma unroll
  for (int k = 0; k < 4; ++k) {
    float xs = dx[k] * w, ys = dy[k] * l;
    cx[k] = x + xs * co - ys * si;
    cy[k] = y + xs * si + ys * co;
  }
}

// Sutherland-Hodgman clip of convex quad 1 by convex quad 2 (both CCW),
// followed by shoelace area. Max 8 vertices after 4 half-plane clips.
__device__ float convex_inter_area(const float* cx1, const float* cy1,
                                   const float* cx2, const float* cy2) {
  float px[8], py[8];
  int n = 4;
#pragma unroll
  for (int k = 0; k < 4; ++k) { px[k] = cx1[k]; py[k] = cy1[k]; }

  for (int e = 0; e < 4; ++e) {
    float ax = cx2[e], ay = cy2[e];
    float bx = cx2[(e + 1) & 3], by = cy2[(e + 1) & 3];
    float ex = bx - ax, ey = by - ay;
    float qx[8], qy[8];
    int m = 0;
    for (int k = 0; k < 8; ++k) {
      if (k >= n) break;
      int k2 = (k + 1 == n) ? 0 : k + 1;
      float s0 = ex * (py[k] - ay) - ey * (px[k] - ax);
      float s1 = ex * (py[k2] - ay) - ey * (px[k2] - ax);
      if (s0 >= 0.0f) { qx[m] = px[k]; qy[m] = py[k]; ++m; }
      if ((s0 > 0.0f && s1 < 0.0f) || (s0 < 0.0f && s1 > 0.0f)) {
        float tt = s0 / (s0 - s1);
        qx[m] = px[k] + tt * (px[k2] - px[k]);
        qy[m] = py[k] + tt * (py[k2] - py[k]);
        ++m;
      }
    }
    n = m;
#pragma unroll
    for (int k = 0; k < 8; ++k) { px[k] = qx[k]; py[k] = qy[k]; }
    if (n == 0) break;
  }

  if (n < 3) return 0.0f;
  float area = 0.0f;
  for (int k = 0; k < 8; ++k) {
    if (k >= n) break;
    int k2 = (k + 1 == n) ? 0 : k + 1;
    area += px[k] * py[k2] - px[k2] * py[k];
  }
  return 0.5f * fabsf(area);
}

__global__ void __launch_bounds__(TPB)
iou3d_loss_partial(const float* __restrict__ pred, const float* __restrict__ target,
                   const float* __restrict__ weight, float* __restrict__ partial, int n) {
  __shared__ float sP[TPB * 7];
  __shared__ float sT[TPB * 7];
  __shared__ float sWave[TPB / 32];

  const int t = threadIdx.x;
  const int base = blockIdx.x * TPB;
  const int totalDw = n * 7;

  // Coalesced async global->LDS staging of the 7-float AoS boxes.
#pragma unroll
  for (int k = 0; k < 7; ++k) {
    int dw = base * 7 + k * TPB + t;
    dw = (dw < totalDw) ? dw : (totalDw - 1);
    unsigned lofP = (unsigned)(uint64_t)&sP[k * TPB + t];
    unsigned lofT = (unsigned)(uint64_t)&sT[k * TPB + t];
    uint64_t gpa = (uint64_t)(pred + dw);
    uint64_t gta = (uint64_t)(target + dw);
    asm volatile("global_load_async_to_lds_b32 %0, %1, off"
                 :: "v"(lofP), "v"(gpa) : "memory");
    asm volatile("global_load_async_to_lds_b32 %0, %1, off"
                 :: "v"(lofT), "v"(gta) : "memory");
  }
  asm volatile("s_wait_asynccnt 0x0" ::: "memory");
  __syncthreads();

  const int i = base + t;
  const bool valid = (i < n);
  const float wgt = weight[valid ? i : (n - 1)];

  const float* bp = &sP[t * 7];  // stride 7 -> bank-conflict-free (gcd(7,64)=1)
  const float* bt = &sT[t * 7];

  float cx1[4], cy1[4], cx2[4], cy2[4];
  bev_corners(bp[0], bp[1], bp[3], bp[4], bp[6], cx1, cy1);
  bev_corners(bt[0], bt[1], bt[3], bt[4], bt[6], cx2, cy2);

  float inter = convex_inter_area(cx1, cy1, cx2, cy2);
  float zmax = fminf(bp[2] + 0.5f * bp[5], bt[2] + 0.5f * bt[5]);
  float zmin = fmaxf(bp[2] - 0.5f * bp[5], bt[2] - 0.5f * bt[5]);
  float ivol = inter * fmaxf(zmax - zmin, 0.0f);
  float v1 = bp[3] * bp[4] * bp[5];
  float v2 = bt[3] * bt[4] * bt[5];
  float iou = ivol / (v1 + v2 - ivol + 1e-8f);
  float contrib = valid ? (1.0f - iou) * wgt : 0.0f;  // select also kills tail NaN

  // ---- Intra-wave reduction on the matrix pipe: D = A(16x4) * ones(4x16) ----
  // A: V0 = per-lane loss, V1 = 0. With B = ones, D[m][n] = lsum(m)+lsum(m+16),
  // replicated over columns. Lanes 0-15 hold rows 0-7 across D's 8 VGPRs,
  // lanes 16-31 hold rows 8-15. EXEC is all-1s here (no early returns above).
  v2f a; a[0] = contrib; a[1] = 0.0f;
  v2f b; b[0] = 1.0f;    b[1] = 1.0f;
  v8f c = {};
  v8f d = __builtin_amdgcn_wmma_f32_16x16x4_f32(false, a, false, b, (short)0, c, false, false);
  float tsum = d[0] + d[1] + d[2] + d[3] + d[4] + d[5] + d[6] + d[7];
  float waveTot = tsum + __shfl_xor(tsum, 16, 32);  // lane0: rows0-7 + rows8-15

  const int wave = t >> 5;
  if ((t & 31) == 0) sWave[wave] = waveTot;
  __syncthreads();
  if (t == 0) {
    float s = 0.0f;
#pragma unroll
    for (int k = 0; k < TPB / 32; ++k) s += sWave[k];
    partial[blockIdx.x] = s;
  }
}

__global__ void __launch_bounds__(TPB)
reduce_mean(const float* __restrict__ partial, int m, float inv_n, float* __restrict__ out) {
  __shared__ float sh[TPB];
  float s = 0.0f;
  for (int i = threadIdx.x; i < m; i += TPB) s += partial[i];
  sh[threadIdx.x] = s;
  __syncthreads();
  for (int off = TPB / 2; off > 0; off >>= 1) {
    if (threadIdx.x < off) sh[threadIdx.x] += sh[threadIdx.x + off];
    __syncthreads();
  }
  if (threadIdx.x == 0) out[0] = sh[0] * inv_n;  // LOSS_WEIGHT * mean, LOSS_WEIGHT = 1
}

extern "C" void kernel_launch(void* const* d_in, const int* in_sizes, int n_in,
                              void* d_out, int out_size, void* d_ws, size_t ws_size,
                              hipStream_t stream) {
  const float* pred   = (const float*)d_in[0];   // (N,7) fp32
  const float* target = (const float*)d_in[1];   // (N,7) fp32
  const float* weight = (const float*)d_in[2];   // (N,)  fp32
  const int n = in_sizes[2];                     // N_BOXES (weight element count)
  float* partial = (float*)d_ws;                 // one fp32 per block (~1954 * 4B)

  const int blocks = (n + TPB - 1) / TPB;
  iou3d_loss_partial<<<blocks, TPB, 0, stream>>>(pred, target, weight, partial, n);
  reduce_mean<<<1, TPB, 0, stream>>>(partial, blocks, 1.0f / (float)n, (float*)d_out);
}